// AttentionFlow_38439957299359
// MI455X (gfx1250) — compile-verified
//
#include <hip/hip_runtime.h>
#include <stdint.h>

#if __has_builtin(__builtin_amdgcn_global_load_async_to_lds_b128)
#define ASYNC_LDS 1
#else
#define ASYNC_LDS 0
#endif

#define RELD 32000  // R(500) * D(64)

typedef int v4i __attribute__((ext_vector_type(4)));
#if ASYNC_LDS
typedef __attribute__((address_space(1))) v4i* gv4i_p;
typedef __attribute__((address_space(3))) v4i* lv4i_p;
#endif

__device__ __forceinline__ unsigned f2mono(float x) {
  unsigned u = __float_as_uint(x);
  return (u & 0x80000000u) ? ~u : (u | 0x80000000u);
}
__device__ __forceinline__ float mono2f(unsigned k) {
  unsigned u = (k & 0x80000000u) ? (k ^ 0x80000000u) : ~k;
  return __uint_as_float(u);
}

// One wave per edge: lanes 0..31 cover d = 2*lane, 2*lane+1.
__global__ __launch_bounds__(256) void k_logits(
    const int* __restrict__ sel, const float* __restrict__ hc,
    const float* __restrict__ hu, const float* __restrict__ rel,
    const float* __restrict__ ws, const float* __restrict__ bvec,
    const float* __restrict__ outw, const float* __restrict__ outb,
    float* __restrict__ logits, unsigned* __restrict__ segmax, int E) {
  __shared__ float s_rel[RELD];  // 128 KB of the 320 KB/WGP LDS

#if ASYNC_LDS
  // ASYNCcnt-tracked DMA of rel_emb into LDS (coalesced b128 per lane).
  for (int i = threadIdx.x * 4; i < RELD; i += blockDim.x * 4) {
    __builtin_amdgcn_global_load_async_to_lds_b128(
        (gv4i_p)(uintptr_t)(rel + i),
        (lv4i_p)(uint32_t)(uintptr_t)(&s_rel[i]),
        0, 0);
  }
  __builtin_amdgcn_s_wait_asynccnt(0);
#else
  for (int i = threadIdx.x; i < RELD; i += blockDim.x) s_rel[i] = rel[i];
#endif
  __syncthreads();

  const int lane = threadIdx.x & 31;
  const int wave = threadIdx.x >> 5;
  const int wavesPerBlock = blockDim.x >> 5;
  const int gwave = blockIdx.x * wavesPerBlock + wave;
  const int nwaves = gridDim.x * wavesPerBlock;
  const int d0 = lane * 2;

  // Per-lane constants (2 channels per lane), hoisted out of the edge loop.
  float2 W[8];
#pragma unroll
  for (int k = 0; k < 8; ++k) W[k] = *(const float2*)(ws + k * 64 + d0);
  const float2 Bv = *(const float2*)(bvec + d0);
  const float2 OW = *(const float2*)(outw + d0);
  const float2 OB = *(const float2*)(outb + d0);

  for (int e = gwave; e < E; e += nwaves) {
    const int4 sA = *(const int4*)(sel + (size_t)e * 8);      // idx, vi, vj, rel
    const int4 sB = *(const int4*)(sel + (size_t)e * 8 + 4);  // idx_vi, idx_vj, e2vi, e2vj
    const int vi = sA.y, vj = sA.z, r = sA.w;
    const int ivi = sB.x, e2vi = sB.z, e2vj = sB.w;

    const float2 A  = *(const float2*)(hc + (size_t)e2vi * 64 + d0);  // hc_vi
    const float2 C  = *(const float2*)(hc + (size_t)e2vj * 64 + d0);  // hc_vj
    const float2 U  = *(const float2*)(hu + (size_t)vi   * 64 + d0);  // hu_vi
    const float2 H  = *(const float2*)(hu + (size_t)vj   * 64 + d0);  // hu_vj
    const float2 Rv = *(const float2*)(&s_rel[r * 64 + d0]);          // rel_emb (LDS)

    // f = b + a*(c*(w0+w1 r) + h*(w2+w3 r)) + u*(c*(w4+w5 r) + h*(w6+w7 r))
    float f0 = Bv.x + A.x * (C.x * fmaf(W[1].x, Rv.x, W[0].x) + H.x * fmaf(W[3].x, Rv.x, W[2].x))
                    + U.x * (C.x * fmaf(W[5].x, Rv.x, W[4].x) + H.x * fmaf(W[7].x, Rv.x, W[6].x));
    float f1 = Bv.y + A.y * (C.y * fmaf(W[1].y, Rv.y, W[0].y) + H.y * fmaf(W[3].y, Rv.y, W[2].y))
                    + U.y * (C.y * fmaf(W[5].y, Rv.y, W[4].y) + H.y * fmaf(W[7].y, Rv.y, W[6].y));
    float g = fmaf(fmaxf(f0, 0.0f), OW.x, OB.x) + fmaf(fmaxf(f1, 0.0f), OW.y, OB.y);

    // wave32 tree reduction over the 64 channels
#pragma unroll
    for (int o = 16; o > 0; o >>= 1) g += __shfl_xor(g, o, 32);

    if (lane == 0) {
      logits[e] = g;
      atomicMax(&segmax[ivi], f2mono(g));
    }
  }
}

__global__ void k_zero(float* out, int n_out, unsigned* segmax, float* segsum, int nseg) {
  int i = blockIdx.x * blockDim.x + threadIdx.x;
  int stride = gridDim.x * blockDim.x;
  for (int t = i; t < n_out; t += stride) out[t] = 0.0f;
  for (int t = i; t < nseg; t += stride) { segmax[t] = 0u; segsum[t] = 0.0f; }
}

__global__ void k_expsum(const int* __restrict__ sel, const float* __restrict__ logits,
                         const unsigned* __restrict__ segmax, float* __restrict__ ebuf,
                         float* __restrict__ segsum, int E) {
  int e = blockIdx.x * blockDim.x + threadIdx.x;
  if (e >= E) return;
  int ivi = sel[(size_t)e * 8 + 4];
  float m = mono2f(segmax[ivi]);
  float ex = expf(logits[e] - m);
  ebuf[e] = ex;
  atomicAdd(&segsum[ivi], ex);
}

__global__ void k_scatter(const int* __restrict__ sel, const float* __restrict__ na,
                          const float* __restrict__ ey, const float* __restrict__ ebuf,
                          const float* __restrict__ segsum, float* __restrict__ out,
                          int E, int N) {
  int e = blockIdx.x * blockDim.x + threadIdx.x;
  if (e >= E) return;
  const int4 sA = *(const int4*)(sel + (size_t)e * 8);
  const int ivi = sel[(size_t)e * 8 + 4];
  const int idx = sA.x, vi = sA.y, vj = sA.z;
  float trans = ebuf[e] / segsum[ivi];
  float ta = na[(size_t)idx * N + vi] * trans * ey[e];
  atomicAdd(&out[(size_t)idx * N + vj], ta);
}

__global__ __launch_bounds__(256) void k_rowsum(const float* __restrict__ out,
                                                float* __restrict__ rowsum, int N) {
  __shared__ float sh[8];
  const int b = blockIdx.x;
  float s = 0.0f;
  for (int i = threadIdx.x; i < N; i += blockDim.x) s += out[(size_t)b * N + i];
#pragma unroll
  for (int o = 16; o > 0; o >>= 1) s += __shfl_xor(s, o, 32);
  if ((threadIdx.x & 31) == 0) sh[threadIdx.x >> 5] = s;
  __syncthreads();
  if (threadIdx.x == 0) {
    float v = 0.0f;
#pragma unroll
    for (int k = 0; k < 8; ++k) v += sh[k];
    rowsum[b] = v;
  }
}

__global__ void k_norm(float* out, const float* __restrict__ rowsum, int BN, int N) {
  int i = blockIdx.x * blockDim.x + threadIdx.x;
  if (i >= BN) return;
  out[i] = out[i] / rowsum[i / N];
}

extern "C" void kernel_launch(void* const* d_in, const int* in_sizes, int n_in,
                              void* d_out, int out_size, void* d_ws, size_t ws_size,
                              hipStream_t stream) {
  const float* na   = (const float*)d_in[0];
  const int*   sel  = (const int*)d_in[1];
  const float* ey   = (const float*)d_in[2];
  const float* hc   = (const float*)d_in[3];
  const float* hu   = (const float*)d_in[4];
  const float* rel  = (const float*)d_in[5];
  const float* ws   = (const float*)d_in[6];
  const float* bvec = (const float*)d_in[7];
  const float* outw = (const float*)d_in[8];
  const float* outb = (const float*)d_in[9];

  const int E  = in_sizes[2];        // edges_y has E elements
  const int BN = in_sizes[0];        // B*N
  const int N  = in_sizes[4] / 64;   // hidden_uncon is 1*N*64
  const int B  = BN / N;

  // Workspace layout: logits[E], ebuf[E], segsum[E], segmax[E] (uint), rowsum[B]
  float*    logits = (float*)d_ws;
  float*    ebuf   = logits + E;
  float*    segsum = ebuf + E;
  unsigned* segmax = (unsigned*)(segsum + E);
  float*    rowsum = (float*)(segmax + E);

  float* out = (float*)d_out;

  k_zero<<<256, 256, 0, stream>>>(out, BN, segmax, segsum, E);
  k_logits<<<512, 256, 0, stream>>>(sel, hc, hu, rel, ws, bvec, outw, outb, logits, segmax, E);
  const int tb = (E + 255) / 256;
  k_expsum<<<tb, 256, 0, stream>>>(sel, logits, segmax, ebuf, segsum, E);
  k_scatter<<<tb, 256, 0, stream>>>(sel, na, ey, ebuf, segsum, out, E, N);
  k_rowsum<<<B, 256, 0, stream>>>(out, rowsum, N);
  k_norm<<<(BN + 255) / 256, 256, 0, stream>>>(out, rowsum, BN, N);
}